// GraphSAGE_10806137716865
// MI455X (gfx1250) — compile-verified
//
#include <hip/hip_runtime.h>

typedef __attribute__((ext_vector_type(2))) float v2f;
typedef __attribute__((ext_vector_type(8))) float v8f;

// ---------------------------------------------------------------------------
// Elementwise / scatter kernels
// ---------------------------------------------------------------------------

__global__ void zero_f32(float* __restrict__ p, long long n) {
    long long i = (long long)blockIdx.x * blockDim.x + threadIdx.x;
    if (i < n) p[i] = 0.0f;
}

__global__ void count_deg(const int* __restrict__ dst, float* __restrict__ deg,
                          long long E) {
    long long e = (long long)blockIdx.x * blockDim.x + threadIdx.x;
    if (e < E) unsafeAtomicAdd(&deg[dst[e]], 1.0f);
}

// sum_e[dst[e], d] += fe[e, d]   (D = 48)
__global__ void scatter_edge(const float* __restrict__ fe, const int* __restrict__ dst,
                             float* __restrict__ sum, long long E) {
    long long t = (long long)blockIdx.x * blockDim.x + threadIdx.x;
    if (t >= E * 48) return;
    int d = (int)(t % 48);
    long long e = t / 48;
    unsafeAtomicAdd(&sum[(long long)dst[e] * 48 + d], fe[t]);
}

// h[n] = concat(fn[n], sum_e[n] / max(deg[n],1))
__global__ void build_h(const float* __restrict__ fn, const float* __restrict__ sum_e,
                        const float* __restrict__ deg, float* __restrict__ h,
                        long long N) {
    long long t = (long long)blockIdx.x * blockDim.x + threadIdx.x;
    if (t >= N * 176) return;
    int c = (int)(t % 176);
    long long n = t / 176;
    float v;
    if (c < 128) v = fn[n * 128 + c];
    else         v = sum_e[n * 48 + (c - 128)] / fmaxf(deg[n], 1.0f);
    h[t] = v;
}

// sum[dst[e], d] += X[src[e], d]
template <int D>
__global__ void scatter_rows(const float* __restrict__ X, const int* __restrict__ src,
                             const int* __restrict__ dst, float* __restrict__ sum,
                             long long E) {
    long long t = (long long)blockIdx.x * blockDim.x + threadIdx.x;
    if (t >= E * D) return;
    int d = (int)(t % D);
    long long e = t / D;
    unsafeAtomicAdd(&sum[(long long)dst[e] * D + d],
                    X[(long long)src[e] * D + d]);
}

template <int D>
__global__ void divide_rows(float* __restrict__ S, const float* __restrict__ deg,
                            long long N) {
    long long t = (long long)blockIdx.x * blockDim.x + threadIdx.x;
    if (t >= N * D) return;
    S[t] = S[t] / fmaxf(deg[t / D], 1.0f);
}

// ---------------------------------------------------------------------------
// Fused SAGE GEMM:  out = act( X@Ws + Xn@Wn + bias )
//   X, Xn : [nrows, K] row-major     Ws, Wn : [K, 256] row-major
// One 512-thread block (16 waves) per 16-row M tile; each wave owns one
// 16-column N tile of the H=256 output and accumulates both products with
// V_WMMA_F32_16X16X4_F32 (full f32 precision; kernel is memory-bound anyway).
// ---------------------------------------------------------------------------
template <int K>
__global__ __launch_bounds__(512) void sage_gemm(
    const float* __restrict__ X, const float* __restrict__ Xn,
    const float* __restrict__ Ws, const float* __restrict__ Wn,
    const float* __restrict__ bias, float* __restrict__ out,
    int nrows, int relu_flag) {
    constexpr int H = 256;
    __shared__ float As[16 * K];
    __shared__ float Ans[16 * K];

    const int tid  = threadIdx.x;
    const int wave = tid >> 5;        // 0..15 -> N tile
    const int lane = tid & 31;
    const int m0   = blockIdx.x * 16; // M tile start

    // Stage the two 16xK A tiles in LDS once per block (shared by all 16 waves)
    for (int i = tid; i < 16 * K; i += 512) {
        int r = i / K;
        int row = m0 + r;
        float a = 0.0f, an = 0.0f;
        if (row < nrows) {
            long long base = (long long)row * K + (i % K);
            a  = X[base];
            an = Xn[base];
        }
        As[i]  = a;
        Ans[i] = an;
    }
    __syncthreads();

    const int col  = (wave << 4) | (lane & 15); // output column 0..255
    const int g    = lane >> 4;                 // lane half -> K-pair select
    const int mrow = lane & 15;                 // A row within tile

    v8f acc = {};
    for (int k = 0; k < K; k += 4) {
        const int kk = k + 2 * g;
        // A 16x4 layout: lane<16 holds K=kk,kk+1 ; lane>=16 holds K=kk(+2),kk+1
        v2f a, an, b, bn;
        a.x  = As[mrow * K + kk];
        a.y  = As[mrow * K + kk + 1];
        an.x = Ans[mrow * K + kk];
        an.y = Ans[mrow * K + kk + 1];
        // B 4x16 layout mirrors C: vgpr0 = rows kk (lanes 0-15) / kk+2 (16-31)
        b.x  = Ws[(long long)kk * H + col];
        b.y  = Ws[(long long)(kk + 1) * H + col];
        bn.x = Wn[(long long)kk * H + col];
        bn.y = Wn[(long long)(kk + 1) * H + col];
        acc = __builtin_amdgcn_wmma_f32_16x16x4_f32(false, a, false, b,
                                                    (short)0, acc, false, false);
        acc = __builtin_amdgcn_wmma_f32_16x16x4_f32(false, an, false, bn,
                                                    (short)0, acc, false, false);
    }

    // C/D layout: vgpr i -> row m0 + i + 8*g, column = col
    const float bv = bias[col];
#pragma unroll
    for (int i = 0; i < 8; ++i) {
        int row = m0 + i + (g << 3);
        if (row < nrows) {
            float v = acc[i] + bv;
            if (relu_flag) v = fmaxf(v, 0.0f);
            out[(long long)row * H + col] = v;
        }
    }
}

// ---------------------------------------------------------------------------
// Host-side orchestration
// ---------------------------------------------------------------------------
extern "C" void kernel_launch(void* const* d_in, const int* in_sizes, int n_in,
                              void* d_out, int out_size, void* d_ws, size_t ws_size,
                              hipStream_t stream) {
    const float* fn  = (const float*)d_in[0]; // [N,128]
    const float* fe  = (const float*)d_in[1]; // [E,48]
    const int*   src = (const int*)d_in[2];   // [E]
    const int*   dst = (const int*)d_in[3];   // [E]
    const float* Ws1 = (const float*)d_in[4]; // [176,256]
    const float* Wn1 = (const float*)d_in[5]; // [176,256]
    const float* b1  = (const float*)d_in[6]; // [256]
    const float* Ws2 = (const float*)d_in[7]; // [256,256]
    const float* Wn2 = (const float*)d_in[8]; // [256,256]
    const float* b2  = (const float*)d_in[9]; // [256]

    const long long N = in_sizes[0] / 128;
    const long long E = in_sizes[2];

    // Scratch layout (floats): deg | sum_e | h | hn1 | h1 | hn2
    float* ws    = (float*)d_ws;
    float* deg   = ws;                 // N
    float* sum_e = deg + N;            // N*48
    float* h     = sum_e + N * 48;     // N*176
    float* hn1   = h + N * 176;        // N*176
    float* h1    = hn1 + N * 176;      // N*256
    float* hn2   = h1 + N * 256;       // N*256

    auto nb = [](long long n) { return (unsigned)((n + 255) / 256); };

    // Stage 0: in-degree + edge-feature mean (deg|sum_e contiguous -> one zero)
    zero_f32<<<nb(N * 49), 256, 0, stream>>>(deg, N * 49);
    count_deg<<<nb(E), 256, 0, stream>>>(dst, deg, E);
    scatter_edge<<<nb(E * 48), 256, 0, stream>>>(fe, dst, sum_e, E);
    build_h<<<nb(N * 176), 256, 0, stream>>>(fn, sum_e, deg, h, N);

    // Stage 1: neighbor mean of h, then fused dual GEMM + ReLU
    zero_f32<<<nb(N * 176), 256, 0, stream>>>(hn1, N * 176);
    scatter_rows<176><<<nb(E * 176), 256, 0, stream>>>(h, src, dst, hn1, E);
    divide_rows<176><<<nb(N * 176), 256, 0, stream>>>(hn1, deg, N);
    const unsigned mtiles = (unsigned)((N + 15) / 16);
    sage_gemm<176><<<mtiles, 512, 0, stream>>>(h, hn1, Ws1, Wn1, b1, h1, (int)N, 1);

    // Stage 2: neighbor mean of h1, then fused dual GEMM (no activation)
    zero_f32<<<nb(N * 256), 256, 0, stream>>>(hn2, N * 256);
    scatter_rows<256><<<nb(E * 256), 256, 0, stream>>>(h1, src, dst, hn2, E);
    divide_rows<256><<<nb(N * 256), 256, 0, stream>>>(hn2, deg, N);
    sage_gemm<256><<<mtiles, 512, 0, stream>>>(h1, hn2, Ws2, Wn2, b2,
                                               (float*)d_out, (int)N, 0);
}